// DecoderForGenerate_57346403336780
// MI455X (gfx1250) — compile-verified
//
#include <hip/hip_runtime.h>
#include <hip/hip_bf16.h>

// Deformable-DETR decoder layer for MI455X (gfx1250, wave32).
// GEMMs: V_WMMA_F32_16X16X4_F32 (fp32-exact) with double-buffered LDS tiles
// filled by GLOBAL_LOAD_ASYNC_TO_LDS_B128 (ASYNCcnt) so the tensor-data path
// overlaps DMA with WMMA issue. value stays [B,S,H,Dh] so sampling gathers
// are 32-lane (128 B) coalesced.

typedef float v2f __attribute__((ext_vector_type(2)));
typedef float v8f __attribute__((ext_vector_type(8)));

#define BDIM 256
#define BM 64
#define BN 64
#define BK 16
#define APAD 20   // As row stride (floats): 80 B, 16B-aligned, conflict-free

#define NB 8
#define NQ 900
#define NC 256
#define NHEADS 8
#define DHEAD 32
#define NS 21760

// Async global->LDS copy of 16 bytes per lane. LDS operand: low 32 bits of the
// generic pointer == LDS byte address (ISA 10.2 aperture truncation).
__device__ __forceinline__ void async_g2l_b128(const float* gsrc, float* ldst)
{
    const unsigned loff = (unsigned)(unsigned long long)(void*)ldst;
    asm volatile("global_load_async_to_lds_b128 %0, %1, off"
                 :: "v"(loff), "v"(gsrc)
                 : "memory");
}
__device__ __forceinline__ void wait_async0()
{
    asm volatile("s_wait_asynccnt 0" ::: "memory");
}

// ---------------------------------------------------------------------------
// Tiled GEMM: C[M,N] = A[M,K] @ B[K,N] + bias, optional relu / row-mask.
// 256 threads = 8 waves; block tile 64x64; wave tile 16x32 (two f32 16x16
// WMMA accumulators). K staged in double-buffered LDS via async-to-LDS DMA.
// flags: bit0 = relu, bit1 = zero rows where mask!=0.
// ---------------------------------------------------------------------------
__global__ __launch_bounds__(BDIM)
void gemm_wmma_f32(const float* __restrict__ A, const float* __restrict__ Bm,
                   const float* __restrict__ bias, float* __restrict__ Cm,
                   int M, int K, int N, int flags,
                   const unsigned char* __restrict__ mask)
{
    __shared__ float As[2][BM][APAD];
    __shared__ float Bs[2][BK][BN];

    const int tid  = threadIdx.x;
    const int wave = tid >> 5;
    const int lane = tid & 31;
    const int l15  = lane & 15;
    const int hi   = lane >> 4;          // 0: K0/K1 half, 1: K2/K3 half

    const int m0 = blockIdx.x * BM;
    const int n0 = blockIdx.y * BN;
    const int wm = (wave >> 1) * 16;     // wave M offset within block
    const int wn = (wave & 1) * 32;      // wave N offset within block

    v8f acc0 = {};
    v8f acc1 = {};

    // staging maps: one 16 B async transfer per thread per tile
    const int a_r = tid >> 2;            // 0..63
    const int a_c = (tid & 3) * 4;       // 0,4,8,12
    const int b_r = tid >> 4;            // 0..15
    const int b_c = (tid & 15) * 4;      // 0..60

    const bool aval = (m0 + a_r) < M;    // stale LDS in OOB rows only feeds
                                         // store-guarded output rows
    const float* aptr = A + (long)(m0 + a_r) * K + a_c;
    const float* bptr = Bm + (long)b_r * N + (n0 + b_c);

    // prologue: stage 0
    if (aval) async_g2l_b128(aptr, &As[0][a_r][a_c]);
    async_g2l_b128(bptr, &Bs[0][b_r][b_c]);
    wait_async0();
    __syncthreads();

    auto compute = [&](int bsel) {
        #pragma unroll
        for (int kk = 0; kk < BK; kk += 4) {
            // A 16x4 frag: lanes 0-15 -> K=kk,kk+1; lanes 16-31 -> K=kk+2,kk+3
            v2f a, b0, b1;
            a.x = As[bsel][wm + l15][kk + 2 * hi + 0];
            a.y = As[bsel][wm + l15][kk + 2 * hi + 1];
            // B 4x16 frag: row = K, lanes stripe N (mirror layout)
            b0.x = Bs[bsel][kk + 2 * hi + 0][wn + l15];
            b0.y = Bs[bsel][kk + 2 * hi + 1][wn + l15];
            b1.x = Bs[bsel][kk + 2 * hi + 0][wn + 16 + l15];
            b1.y = Bs[bsel][kk + 2 * hi + 1][wn + 16 + l15];
            acc0 = __builtin_amdgcn_wmma_f32_16x16x4_f32(
                false, a, false, b0, (short)0, acc0, false, false);
            acc1 = __builtin_amdgcn_wmma_f32_16x16x4_f32(
                false, a, false, b1, (short)0, acc1, false, false);
        }
    };

    int buf = 0;
    for (int k0 = BK; k0 < K; k0 += BK) {
        const int nb = buf ^ 1;
        // kick off next stage DMA, then overlap with this stage's WMMAs
        if (aval) async_g2l_b128(aptr + k0, &As[nb][a_r][a_c]);
        async_g2l_b128(bptr + (long)k0 * N, &Bs[nb][b_r][b_c]);
        compute(buf);
        wait_async0();
        __syncthreads();
        buf = nb;
    }
    compute(buf);

    // epilogue: C/D layout -> lane l, vgpr g holds (M = g + 8*hi, N = l15)
    const float bias0 = bias ? bias[n0 + wn + l15] : 0.f;
    const float bias1 = bias ? bias[n0 + wn + 16 + l15] : 0.f;
    #pragma unroll
    for (int g = 0; g < 8; ++g) {
        const int row = m0 + wm + g + hi * 8;
        if (row < M) {
            float v0 = acc0[g] + bias0;
            float v1 = acc1[g] + bias1;
            if (flags & 1) { v0 = fmaxf(v0, 0.f); v1 = fmaxf(v1, 0.f); }
            if (flags & 2) { if (mask[row]) { v0 = 0.f; v1 = 0.f; } }
            Cm[(long)row * N + n0 + wn + l15]      = v0;
            Cm[(long)row * N + n0 + wn + 16 + l15] = v1;
        }
    }
}

// ---------------------------------------------------------------------------
// Softmax over rows of 64 (one wave per row; lane holds elements l and l+32).
// ---------------------------------------------------------------------------
__global__ __launch_bounds__(BDIM)
void softmax64(float* __restrict__ attn, int rows)
{
    const int row  = blockIdx.x * 8 + (threadIdx.x >> 5);
    const int lane = threadIdx.x & 31;
    if (row >= rows) return;
    float* p = attn + (long)row * 64;
    float v0 = p[lane], v1 = p[lane + 32];
    float m = fmaxf(v0, v1);
    #pragma unroll
    for (int o = 16; o > 0; o >>= 1) m = fmaxf(m, __shfl_xor(m, o, 32));
    float e0 = expf(v0 - m), e1 = expf(v1 - m);
    float s = e0 + e1;
    #pragma unroll
    for (int o = 16; o > 0; o >>= 1) s += __shfl_xor(s, o, 32);
    const float inv = 1.f / s;
    p[lane]      = e0 * inv;
    p[lane + 32] = e1 * inv;
}

// ---------------------------------------------------------------------------
// Multi-scale deformable sampling. One wave per (b,q,h); lane = channel d.
// Gathers are 32 consecutive floats (128 B) per corner -> fully coalesced.
// ---------------------------------------------------------------------------
__global__ __launch_bounds__(BDIM)
void ms_sample(const float* __restrict__ value,   // [B*S, 256] (b,s,h,dh)
               const float* __restrict__ off,     // [B*Q, 1024]
               const float* __restrict__ attn,    // [B*Q, 512] (softmaxed)
               const float* __restrict__ pred,    // [B,Q,4]
               const float* __restrict__ vr,      // [B,4,2]
               float* __restrict__ out)           // [B*Q, 256]
{
    const int wid  = blockIdx.x * 8 + (threadIdx.x >> 5);  // bq*8 + h
    const int lane = threadIdx.x & 31;
    const int h  = wid & 7;
    const int bq = wid >> 3;
    const int b  = bq / NQ;

    const float px = pred[bq * 4 + 0], py = pred[bq * 4 + 1];
    const float pw = pred[bq * 4 + 2], ph = pred[bq * 4 + 3];
    const float* offp  = off  + (long)bq * 1024 + h * 128;
    const float* attnp = attn + (long)bq * 512  + h * 64;

    const int HWl[4][3] = {{128,128,0},{64,64,16384},{32,32,20480},{16,16,21504}};
    float acc = 0.f;

    #pragma unroll
    for (int l = 0; l < 4; ++l) {
        const int Hl = HWl[l][0], Wl = HWl[l][1], s0 = HWl[l][2];
        const float vrx = vr[(b * 4 + l) * 2 + 0];
        const float vry = vr[(b * 4 + l) * 2 + 1];
        const float cx = px * vrx, cy = py * vry;
        const float bw = pw * vrx, bh = ph * vry;
        const float* vbase = value + ((long)(b * NS + s0)) * NC + h * DHEAD + lane;

        for (int p = 0; p < 16; ++p) {
            const float ox = offp[l * 32 + p * 2 + 0];
            const float oy = offp[l * 32 + p * 2 + 1];
            const float lx = cx + ox * (1.0f / 16.0f) * bw * 0.5f;
            const float ly = cy + oy * (1.0f / 16.0f) * bh * 0.5f;
            const float x = lx * (float)Wl - 0.5f;
            const float y = ly * (float)Hl - 0.5f;
            const float xf = floorf(x), yf = floorf(y);
            const int x0 = (int)xf, y0 = (int)yf;
            const float wx = x - xf, wy = y - yf;
            const float aw  = attnp[l * 16 + p];
            const float w00 = (1.f - wx) * (1.f - wy) * aw;
            const float w01 = wx * (1.f - wy) * aw;
            const float w10 = (1.f - wx) * wy * aw;
            const float w11 = wx * wy * aw;
            float v00 = 0.f, v01 = 0.f, v10 = 0.f, v11 = 0.f;
            const bool ix0 = (x0 >= 0) & (x0 < Wl);
            const bool ix1 = (x0 + 1 >= 0) & (x0 + 1 < Wl);
            const bool iy0 = (y0 >= 0) & (y0 < Hl);
            const bool iy1 = (y0 + 1 >= 0) & (y0 + 1 < Hl);
            if (ix0 & iy0) v00 = vbase[(long)(y0 * Wl + x0) * NC];
            if (ix1 & iy0) v01 = vbase[(long)(y0 * Wl + x0 + 1) * NC];
            if (ix0 & iy1) v10 = vbase[(long)((y0 + 1) * Wl + x0) * NC];
            if (ix1 & iy1) v11 = vbase[(long)((y0 + 1) * Wl + x0 + 1) * NC];
            acc += w00 * v00 + w01 * v01 + w10 * v10 + w11 * v11;
        }
    }
    out[(long)bq * NC + h * DHEAD + lane] = acc;
}

// ---------------------------------------------------------------------------
// out = LayerNorm(a + r) * g + b.  One wave per row of 256 (8 vals/lane).
// ---------------------------------------------------------------------------
__global__ __launch_bounds__(BDIM)
void add_layernorm(const float* __restrict__ a, const float* __restrict__ r,
                   const float* __restrict__ g, const float* __restrict__ be,
                   float* __restrict__ out, int rows)
{
    const int row  = blockIdx.x * 8 + (threadIdx.x >> 5);
    const int lane = threadIdx.x & 31;
    if (row >= rows) return;
    const float* pa = a + (long)row * NC;
    const float* pr = r + (long)row * NC;
    float v[8];
    float s = 0.f;
    #pragma unroll
    for (int i = 0; i < 8; ++i) {
        v[i] = pa[lane + i * 32] + pr[lane + i * 32];
        s += v[i];
    }
    #pragma unroll
    for (int o = 16; o > 0; o >>= 1) s += __shfl_xor(s, o, 32);
    const float mu = s * (1.f / 256.f);
    float var = 0.f;
    #pragma unroll
    for (int i = 0; i < 8; ++i) { const float d = v[i] - mu; var += d * d; }
    #pragma unroll
    for (int o = 16; o > 0; o >>= 1) var += __shfl_xor(var, o, 32);
    var *= (1.f / 256.f);
    const float rs = rsqrtf(var + 1e-5f);
    #pragma unroll
    for (int i = 0; i < 8; ++i) {
        const int c = lane + i * 32;
        out[(long)row * NC + c] = (v[i] - mu) * rs * g[c] + be[c];
    }
}

// ---------------------------------------------------------------------------
extern "C" void kernel_launch(void* const* d_in, const int* in_sizes, int n_in,
                              void* d_out, int out_size, void* d_ws, size_t ws_size,
                              hipStream_t stream)
{
    (void)in_sizes; (void)n_in; (void)out_size; (void)ws_size;

    const float* query  = (const float*)d_in[0];
    const float* pred   = (const float*)d_in[1];
    const float* memory = (const float*)d_in[2];
    const unsigned char* mask = (const unsigned char*)d_in[3];
    const float* vr   = (const float*)d_in[6];
    const float* Wv   = (const float*)d_in[7];  const float* bv   = (const float*)d_in[8];
    const float* Woff = (const float*)d_in[9];  const float* boff = (const float*)d_in[10];
    const float* Wa   = (const float*)d_in[11]; const float* ba   = (const float*)d_in[12];
    const float* Wo   = (const float*)d_in[13]; const float* bo   = (const float*)d_in[14];
    const float* gca  = (const float*)d_in[15]; const float* bca  = (const float*)d_in[16];
    const float* W1   = (const float*)d_in[17]; const float* b1   = (const float*)d_in[18];
    const float* W2   = (const float*)d_in[19]; const float* b2   = (const float*)d_in[20];
    const float* g3   = (const float*)d_in[21]; const float* b3   = (const float*)d_in[22];
    float* out = (float*)d_out;

    const long M_VAL = (long)NB * NS;     // 174080
    const long M_Q   = (long)NB * NQ;     // 7200

    float* ws      = (float*)d_ws;
    float* w_value = ws;                               // 174080*256
    float* w_off   = w_value + M_VAL * NC;             // 7200*1024
    float* w_attn  = w_off   + M_Q * 1024;             // 7200*512
    float* w_samp  = w_attn  + M_Q * 512;              // 7200*256
    float* w_tgt2  = w_samp  + M_Q * NC;               // 7200*256
    float* w_x     = w_tgt2  + M_Q * NC;               // 7200*256
    float* w_h     = w_x     + M_Q * NC;               // 7200*2048
    float* w_f     = w_h     + M_Q * 2048;             // 7200*256

    const int gmq = (int)((M_Q + BM - 1) / BM);        // 113

    // value = memory @ Wv + bv, masked rows zeroed
    gemm_wmma_f32<<<dim3((int)(M_VAL / BM), NC / BN), BDIM, 0, stream>>>(
        memory, Wv, bv, w_value, (int)M_VAL, NC, NC, 2, mask);
    // sampling offsets and attention logits
    gemm_wmma_f32<<<dim3(gmq, 1024 / BN), BDIM, 0, stream>>>(
        query, Woff, boff, w_off, (int)M_Q, NC, 1024, 0, nullptr);
    gemm_wmma_f32<<<dim3(gmq, 512 / BN), BDIM, 0, stream>>>(
        query, Wa, ba, w_attn, (int)M_Q, NC, 512, 0, nullptr);
    softmax64<<<(int)(M_Q * NHEADS / 8), BDIM, 0, stream>>>(w_attn, (int)(M_Q * NHEADS));
    // bilinear sampling + attention weighting
    ms_sample<<<(int)(M_Q * NHEADS / 8), BDIM, 0, stream>>>(
        w_value, w_off, w_attn, pred, vr, w_samp);
    // output projection, residual + LN
    gemm_wmma_f32<<<dim3(gmq, NC / BN), BDIM, 0, stream>>>(
        w_samp, Wo, bo, w_tgt2, (int)M_Q, NC, NC, 0, nullptr);
    add_layernorm<<<(int)(M_Q / 8), BDIM, 0, stream>>>(query, w_tgt2, gca, bca, w_x, (int)M_Q);
    // FFN
    gemm_wmma_f32<<<dim3(gmq, 2048 / BN), BDIM, 0, stream>>>(
        w_x, W1, b1, w_h, (int)M_Q, NC, 2048, 1, nullptr);
    gemm_wmma_f32<<<dim3(gmq, NC / BN), BDIM, 0, stream>>>(
        w_h, W2, b2, w_f, (int)M_Q, 2048, NC, 0, nullptr);
    add_layernorm<<<(int)(M_Q / 8), BDIM, 0, stream>>>(w_x, w_f, g3, b3, out, (int)M_Q);
}